// PeriodicPrimitives2D_88072599372026
// MI455X (gfx1250) — compile-verified
//
#include <hip/hip_runtime.h>
#include <hip/hip_bf16.h>

// ---------------------------------------------------------------------------
// PeriodicPrimitives2D forward for gfx1250 (MI455X).
//   out[n,c] = sum_g  exp(-0.5*((rx*s0)^2+(ry*s1)^2))
//              * prod_d sum_k cf[g,d,k] * cos(2pi * x[n,d] * f[g,d,k])
//              * colors[g,c]
// VALU/transcendental-bound. Per 256-gaussian chunk:
//   - TDM (tensor_load_to_lds) double-buffers the raw parameter arrays into
//     LDS while the previous chunk is being computed (TENSORcnt-synced).
//   - One staging pass derives per-gaussian affine constants in LDS.
//   - Weights use native v_cos/v_exp (revolution-domain inputs).
//   - The [16x4]x[4x3] color contraction runs on V_WMMA_F32_16X16X4_F32.
// ---------------------------------------------------------------------------

typedef __attribute__((ext_vector_type(2))) float v2f;
typedef __attribute__((ext_vector_type(8))) float v8f;
typedef unsigned int u32;
typedef __attribute__((ext_vector_type(4))) unsigned int u32x4;
typedef __attribute__((ext_vector_type(4))) int i32x4;
typedef __attribute__((ext_vector_type(8))) int i32x8;

#define GCHUNK 256          // gaussians staged per chunk
#define PFLOATS 24          // padded floats per gaussian (96B stride, 16B aligned)
#define WAVES_PER_BLOCK 8
#define PTS_PER_WAVE 64     // 4 WMMA row tiles of 16 points
#define PTS_PER_BLOCK (WAVES_PER_BLOCK * PTS_PER_WAVE)  // 512

// -------- LDS layout (single __shared__ block, byte offsets) --------
#define OFF_RCOEF 0                       // 2048 f32  (8 KB) raw coeffs
#define OFF_RIDX  8192                    // 2048 i32  (8 KB) raw indices
#define OFF_RPOS  16384                   //  512 f32  (2 KB) raw positions
#define OFF_RSCL  18432                   //  512 f32  (2 KB) raw scales
#define OFF_RROT  20480                   //  256 f32  (1 KB) raw rotations
#define OFF_RCOL  21504                   //  768 f32  (3 KB) raw colors
#define OFF_PAR   24576                   // 6144 f32 (24 KB) derived params
#define OFF_COL   49152                   // 1024 f32  (4 KB) padded colors
#define LDS_BYTES 53248                   // 52 KB total

// cos(2*pi*t): v_cos_f32 input is in revolutions; fract keeps it in-range.
__device__ __forceinline__ float cos2pi(float t) {
  return __builtin_amdgcn_cosf(__builtin_amdgcn_fractf(t));
}

// Flat shared pointer -> LDS byte offset (ISA: LDS aperture uses addr[31:0]).
__device__ __forceinline__ u32 lds_off(const void* p) {
  return (u32)(uintptr_t)p;
}

// Issue one 1-D TDM copy of `ndw` dwords: global (gptr) -> LDS (ldsp).
// D# built per CDNA5 ISA: 2D tile, tile_dim0 = ndw,
// tile_dim1 = 0 (unused), data_size = 4 bytes, type = 2, no cluster multicast.
__device__ __forceinline__ void tdm_load_1d(const void* gptr, const void* ldsp,
                                            u32 ndw) {
  const u32 galo = (u32)(uintptr_t)gptr;
  const u32 gahi = (u32)(((uintptr_t)gptr) >> 32);
  u32x4 g0;
  g0.x = 1u;                                   // count=1, user mode, no gather
  g0.y = lds_off(ldsp);                        // lds_addr (bytes)
  g0.z = galo;                                 // global_addr[31:0]
  g0.w = (gahi & 0x1FFFFFFu) | (2u << 30);     // global_addr[56:32] | type=2
  i32x8 g1;
  g1[0] = (int)(2u << 16);                     // workgroup_mask=0, data_size=4B
  g1[1] = (int)((ndw & 0xFFFFu) << 16);        // tensor_dim0[15:0]
  g1[2] = (int)(((ndw >> 16) & 0xFFFFu) | (1u << 16)); // dim0[31:16], dim1=1
  g1[3] = (int)((ndw & 0xFFFFu) << 16);        // tile_dim0 = ndw
  g1[4] = 0;                                   // tile_dim1/2 = 0 (unused)
  g1[5] = (int)ndw;                            // tensor_dim0_stride[31:0]
  g1[6] = 0;
  g1[7] = 0;
  const i32x4 z4 = {0, 0, 0, 0};
#if __clang_major__ >= 23
  const i32x8 z8 = {0, 0, 0, 0, 0, 0, 0, 0};
  __builtin_amdgcn_tensor_load_to_lds(g0, g1, z4, z4, z8, 0);
#else
  __builtin_amdgcn_tensor_load_to_lds(g0, g1, z4, z4, 0);
#endif
}

// Weight for one (gaussian, point) pair from the staged param vectors.
//  P0 = (a0, a1, b0, c0)        P1 = (c1, b1, f00, f01)
//  P2 = (f02, f03, cf00, cf01)  P3 = (cf02, cf03, f10, f11)
//  P4 = (f12, f13, cf10, cf11)  P5 = (cf12, cf13, 0, 0)
__device__ __forceinline__ float gweight(float4 P0, float4 P1, float4 P2,
                                         float4 P3, float4 P4, float4 P5,
                                         float x0, float x1) {
  float u = fmaf(P0.x, x0, fmaf(P0.y, x1, P0.z));   // rx*s0
  float v = fmaf(P0.w, x0, fmaf(P1.x, x1, P1.y));   // ry*s1
  float tt = fmaf(u, u, v * v);
  float e = __builtin_amdgcn_exp2f(tt * -0.7213475204444817f); // exp(-0.5*tt)
  float s0 = P2.z * cos2pi(x0 * P1.z);
  s0 = fmaf(P2.w, cos2pi(x0 * P1.w), s0);
  s0 = fmaf(P3.x, cos2pi(x0 * P2.x), s0);
  s0 = fmaf(P3.y, cos2pi(x0 * P2.y), s0);
  float s1 = P4.z * cos2pi(x1 * P3.z);
  s1 = fmaf(P4.w, cos2pi(x1 * P3.w), s1);
  s1 = fmaf(P5.x, cos2pi(x1 * P4.x), s1);
  s1 = fmaf(P5.y, cos2pi(x1 * P4.y), s1);
  return e * s0 * s1;
}

__global__ __launch_bounds__(256) void pp2d_kernel(
    const float* __restrict__ x,           // [N,2]
    const float* __restrict__ colors,      // [G,3]
    const float* __restrict__ positions,   // [G,2]
    const float* __restrict__ scales,      // [G,2]
    const float* __restrict__ rotations,   // [G,1]
    const float* __restrict__ coeffs,      // [G,2,4]
    const int*  __restrict__ windices,     // [G,2,4]
    const int*  __restrict__ nfreq_p,      // scalar
    const int*  __restrict__ maxf_p,       // scalar
    float* __restrict__ out,               // [N,3]
    int N, int G) {
  __shared__ alignas(16) unsigned char s_lds[LDS_BYTES];
  float* rawCoef = (float*)(s_lds + OFF_RCOEF);
  int*   rawIdx  = (int*)  (s_lds + OFF_RIDX);
  float* rawPos  = (float*)(s_lds + OFF_RPOS);
  float* rawScl  = (float*)(s_lds + OFF_RSCL);
  float* rawRot  = (float*)(s_lds + OFF_RROT);
  float* rawCol  = (float*)(s_lds + OFF_RCOL);
  float* s_par   = (float*)(s_lds + OFF_PAR);   // GCHUNK * PFLOATS
  float* s_col   = (float*)(s_lds + OFF_COL);   // GCHUNK * 4 (slot 3 zeroed)

  const int t    = threadIdx.x;
  const int lane = t & 31;
  const int wave = t >> 5;
  const int colN = lane & 15;            // WMMA column / row-within-tile
  const int kk0  = (lane >> 4) << 1;     // 0 for lanes 0-15, 2 for lanes 16-31
  const int pBase = blockIdx.x * PTS_PER_BLOCK + wave * PTS_PER_WAVE;

  const float fscale = (float)maxf_p[0] / (float)nfreq_p[0];

  // Per-lane point coordinates for the 4 row tiles this wave owns.
  float x0v[4], x1v[4];
#pragma unroll
  for (int r = 0; r < 4; ++r) {
    int p = pBase + r * 16 + colN;
    p = p < N ? p : N - 1;               // clamp (stores are guarded later)
    x0v[r] = x[p * 2 + 0];
    x1v[r] = x[p * 2 + 1];
  }

  v8f acc[4];
#pragma unroll
  for (int r = 0; r < 4; ++r)
    acc[r] = (v8f){0.f, 0.f, 0.f, 0.f, 0.f, 0.f, 0.f, 0.f};

  const int nchunks = (G + GCHUNK - 1) / GCHUNK;

  // Kick off the TDM pipeline: raw arrays of chunk 0 -> LDS (wave 0 only).
  if (wave == 0) {
    const u32 rem = (u32)((G < GCHUNK) ? G : GCHUNK);
    tdm_load_1d(coeffs,    rawCoef, rem * 8u);
    tdm_load_1d(windices,  rawIdx,  rem * 8u);
    tdm_load_1d(positions, rawPos,  rem * 2u);
    tdm_load_1d(scales,    rawScl,  rem * 2u);
    tdm_load_1d(rotations, rawRot,  rem);
    tdm_load_1d(colors,    rawCol,  rem * 3u);
  }

  for (int ch = 0; ch < nchunks; ++ch) {
    const int gbase = ch * GCHUNK;
    // Wait for this chunk's TDM copies, then publish to the workgroup.
    if (wave == 0) __builtin_amdgcn_s_wait_tensorcnt(0);
    __syncthreads();   // raw chunk visible; previous chunk's s_par readers done

    // ---- stage: thread t derives params of gaussian gbase+t from LDS raw ----
    {
      const int g = gbase + t;
      float a0 = 0.f, a1 = 0.f, b0 = 0.f, c0 = 0.f, c1 = 0.f, b1 = 0.f;
      float f[2][4], cf[2][4];
      float c3[3] = {0.f, 0.f, 0.f};
#pragma unroll
      for (int d = 0; d < 2; ++d)
#pragma unroll
        for (int k = 0; k < 4; ++k) { f[d][k] = 0.f; cf[d][k] = 0.f; }
      if (g < G) {
        float px = rawPos[t * 2 + 0], py = rawPos[t * 2 + 1];
        float s0 = rawScl[t * 2 + 0], s1 = rawScl[t * 2 + 1];
        float th = rawRot[t];
        const float INV2PI = 0.15915494309189535f;
        float ct = __builtin_amdgcn_cosf(th * INV2PI);
        float st = __builtin_amdgcn_sinf(th * INV2PI);
        a0 = s0 * ct;  a1 = s0 * st;  b0 = -(a0 * px + a1 * py);
        c0 = -s1 * st; c1 = s1 * ct;  b1 = -(c0 * px + c1 * py);
#pragma unroll
        for (int d = 0; d < 2; ++d)
#pragma unroll
          for (int k = 0; k < 4; ++k) {
            cf[d][k] = rawCoef[t * 8 + d * 4 + k];
            f[d][k]  = fscale * (float)rawIdx[t * 8 + d * 4 + k];
          }
        c3[0] = rawCol[t * 3 + 0];
        c3[1] = rawCol[t * 3 + 1];
        c3[2] = rawCol[t * 3 + 2];
      }
      float* P = &s_par[t * PFLOATS];
      P[0]  = a0;       P[1]  = a1;       P[2]  = b0;       P[3]  = c0;
      P[4]  = c1;       P[5]  = b1;       P[6]  = f[0][0];  P[7]  = f[0][1];
      P[8]  = f[0][2];  P[9]  = f[0][3];  P[10] = cf[0][0]; P[11] = cf[0][1];
      P[12] = cf[0][2]; P[13] = cf[0][3]; P[14] = f[1][0];  P[15] = f[1][1];
      P[16] = f[1][2];  P[17] = f[1][3];  P[18] = cf[1][0]; P[19] = cf[1][1];
      P[20] = cf[1][2]; P[21] = cf[1][3]; P[22] = 0.f;      P[23] = 0.f;
      s_col[t * 4 + 0] = c3[0];
      s_col[t * 4 + 1] = c3[1];
      s_col[t * 4 + 2] = c3[2];
      s_col[t * 4 + 3] = 0.f;   // zero pad -> B columns >= 3 read zero
    }
    __syncthreads();   // raw buffers free to be overwritten from here on

    // Issue TDM for the next chunk; copies overlap this chunk's compute.
    if (wave == 0 && (ch + 1) < nchunks) {
      const int nb = gbase + GCHUNK;
      const u32 rem = (u32)(((G - nb) < GCHUNK) ? (G - nb) : GCHUNK);
      tdm_load_1d(coeffs + (size_t)nb * 8,   rawCoef, rem * 8u);
      tdm_load_1d(windices + (size_t)nb * 8, rawIdx,  rem * 8u);
      tdm_load_1d(positions + (size_t)nb * 2, rawPos, rem * 2u);
      tdm_load_1d(scales + (size_t)nb * 2,   rawScl,  rem * 2u);
      tdm_load_1d(rotations + (size_t)nb,    rawRot,  rem);
      tdm_load_1d(colors + (size_t)nb * 3,   rawCol,  rem * 3u);
    }

    // ---- compute: 64 groups of 4 gaussians, one WMMA per row tile ----
    const float4* lds4 = (const float4*)s_par;
    for (int grp = 0; grp < GCHUNK / 4; ++grp) {
      const int gA = grp * 4 + kk0;      // this lane's first gaussian (A VGPR0)
      float wA[4], wB[4];
      {
        const float4* P = &lds4[gA * (PFLOATS / 4)];
        float4 P0 = P[0], P1 = P[1], P2 = P[2], P3 = P[3], P4 = P[4], P5 = P[5];
#pragma unroll
        for (int r = 0; r < 4; ++r)
          wA[r] = gweight(P0, P1, P2, P3, P4, P5, x0v[r], x1v[r]);
      }
      {
        const float4* P = &lds4[(gA + 1) * (PFLOATS / 4)];
        float4 P0 = P[0], P1 = P[1], P2 = P[2], P3 = P[3], P4 = P[4], P5 = P[5];
#pragma unroll
        for (int r = 0; r < 4; ++r)
          wB[r] = gweight(P0, P1, P2, P3, P4, P5, x0v[r], x1v[r]);
      }
      // B tile (4x16 colors): K rows split lane<16 -> K=0/1, lane>=16 -> K=2/3,
      // mirroring the documented 16x4 A layout. Column = colN, zero for >= 3.
      const int mc = colN < 3 ? colN : 3;       // slot 3 holds 0.0
      v2f b;
      b.x = s_col[gA * 4 + mc];
      b.y = s_col[(gA + 1) * 4 + mc];
#pragma unroll
      for (int r = 0; r < 4; ++r) {
        v2f a;
        a.x = wA[r];
        a.y = wB[r];
        acc[r] = __builtin_amdgcn_wmma_f32_16x16x4_f32(
            /*neg_a=*/false, a, /*neg_b=*/false, b,
            /*c_mod=*/(short)0, acc[r], /*reuse_a=*/false, /*reuse_b=*/false);
      }
    }
  }

  // ---- store: D tile VGPR j = row M=j (lanes 0-15) / M=8+j (lanes 16-31) ----
  const int hcol = lane >> 4;
  if (colN < 3) {
#pragma unroll
    for (int r = 0; r < 4; ++r) {
      const int prow = pBase + r * 16 + hcol * 8;
#pragma unroll
      for (int j = 0; j < 8; ++j) {
        const int p = prow + j;
        if (p < N) out[p * 3 + colN] = acc[r][j];
      }
    }
  }
}

extern "C" void kernel_launch(void* const* d_in, const int* in_sizes, int n_in,
                              void* d_out, int out_size, void* d_ws, size_t ws_size,
                              hipStream_t stream) {
  const float* x         = (const float*)d_in[0];
  const float* colors    = (const float*)d_in[1];
  const float* positions = (const float*)d_in[2];
  const float* scales    = (const float*)d_in[3];
  const float* rotations = (const float*)d_in[4];
  const float* coeffs    = (const float*)d_in[5];
  const int*   windices  = (const int*)d_in[6];
  const int*   nfreq     = (const int*)d_in[7];
  const int*   maxf      = (const int*)d_in[8];
  float* out = (float*)d_out;

  const int N = in_sizes[0] / 2;   // 65536
  const int G = in_sizes[1] / 3;   // 4096

  const int grid = (N + PTS_PER_BLOCK - 1) / PTS_PER_BLOCK;  // 128 blocks
  pp2d_kernel<<<grid, 256, 0, stream>>>(x, colors, positions, scales, rotations,
                                        coeffs, windices, nfreq, maxf, out, N, G);
}